// RelativePolarCoordPredictor_30219389895181
// MI455X (gfx1250) — compile-verified
//
#include <hip/hip_runtime.h>
#include <math.h>
#include <stdint.h>

#define N_BATCH  256
#define C_CH     2048
#define HW       49
#define SZ       7
#define TILE_C   32
#define NTILES   (C_CH / TILE_C)    // 64
#define NTHREADS 512                // 16 wave32 waves
#define NSTRIPES 8                  // 512 threads / 64 p-slots
#define RING     9                  // LDS ring slots
#define DEPTH    8                  // tiles issued ahead
#define TILE_F   (TILE_C * HW)      // 1568 floats per slot
#define TILE_B   (TILE_F * 4)       // 6272 bytes per slot (= 392 x b128)
#define RING_B   (RING * TILE_B)    // 56448 bytes
#define NV16     (TILE_B / 16)      // 392 b128 transfers per tile

typedef float v4f __attribute__((ext_vector_type(4)));

// Low 32 bits of a generic pointer to __shared__ == LDS byte address.
__device__ __forceinline__ uint32_t lds_u32(const void* p) {
    return (uint32_t)(uintptr_t)p;
}

__global__ __launch_bounds__(NTHREADS)
void rpcp_kernel(const float* __restrict__ x, const float* __restrict__ Wlin,
                 const float* __restrict__ bias, float* __restrict__ out) {
    __shared__ alignas(16) float ring[RING * TILE_F];   // 56448 B
    __shared__ int s_idx;

    const int tid    = threadIdx.x;
    const int lane   = tid & 31;
    const int wave   = tid >> 5;      // 0..15
    const int p      = tid & 63;      // position slot (valid when < 49)
    const int stripe = tid >> 6;      // 0..7 channel stripes (wave-uniform)
    const int n      = blockIdx.x;

    const float* xn = x + (size_t)n * C_CH * HW;   // contiguous (C,49) slab

    // ---- flat-memcpy DMA: one async b128 per wave per tile ----
    const int  E          = wave * 32 + lane;       // 16B-chunk id, 0..511
    const bool dma_active = (E < NV16);             // waves 13..15 idle (cnt==0)
    const float*   gp     = xn + (size_t)E * 4;     // global, +TILE_F floats/tile
    const uint32_t lbase  = lds_u32(ring);
    uint32_t       lp     = lbase + (uint32_t)E * 16u;  // LDS, +TILE_B/tile, wraps

    auto issue_tile = [&]() {
        if (dma_active) {
            asm volatile("global_load_async_to_lds_b128 %0, %1, off"
                         :: "v"(lp), "v"(gp) : "memory");
        }
        gp += TILE_F;
        lp += TILE_B;
        if (lp >= lbase + RING_B) lp -= RING_B;
    };

    float a0 = 0.f, a1 = 0.f, a2 = 0.f, a3 = 0.f, a4 = 0.f;

#pragma unroll
    for (int t = 0; t < DEPTH; ++t) issue_tile();   // fill pipeline 8 deep

    int cslot = 0;
    for (int t = 0; t < NTILES; ++t) {
        if (t + DEPTH < NTILES) issue_tile();

        // Scalar-cache weight fetch for this tile's 4 channels (wave-uniform,
        // 16B aligned). Issued early so KMcnt latency hides under the wait.
        v4f w10, w11, w20, w21;
        {
            const int c0 = __builtin_amdgcn_readfirstlane(t * TILE_C + stripe * 4);
            const float* wp = Wlin + c0;
            asm volatile("s_load_b128 %0, %4, 0x0\n\t"     // W1[0][c0..c0+3]
                         "s_load_b128 %1, %4, 0x4000\n\t"  // W1[1]
                         "s_load_b128 %2, %4, 0x2000\n\t"  // W2[0]
                         "s_load_b128 %3, %4, 0x6000"      // W2[1]
                         : "=s"(w10), "=s"(w11), "=s"(w20), "=s"(w21)
                         : "s"(wp));
        }

        // Tile t is the oldest in the per-wave async FIFO: force the counter
        // down to (#tiles outstanding beyond t) x 1 issue.
        const int rem0 = NTILES - 1 - t;
        const int rem  = rem0 < DEPTH ? rem0 : DEPTH;
        switch (rem) {
            case 8: asm volatile("s_wait_asynccnt 8" ::: "memory"); break;
            case 7: asm volatile("s_wait_asynccnt 7" ::: "memory"); break;
            case 6: asm volatile("s_wait_asynccnt 6" ::: "memory"); break;
            case 5: asm volatile("s_wait_asynccnt 5" ::: "memory"); break;
            case 4: asm volatile("s_wait_asynccnt 4" ::: "memory"); break;
            case 3: asm volatile("s_wait_asynccnt 3" ::: "memory"); break;
            case 2: asm volatile("s_wait_asynccnt 2" ::: "memory"); break;
            case 1: asm volatile("s_wait_asynccnt 1" ::: "memory"); break;
            default: asm volatile("s_wait_asynccnt 0" ::: "memory"); break;
        }
        __syncthreads();                            // all waves' DMA visible

        // Wait for the scalar loads; tied "+s" operands make the FMAs depend
        // on this asm so they cannot be hoisted above the wait.
        asm volatile("s_wait_kmcnt 0x0"
                     : "+s"(w10), "+s"(w11), "+s"(w20), "+s"(w21));

        if (p < HW) {
            const float* tb = &ring[cslot * TILE_F];
            const int cb = stripe * 4;
            const float v0 = tb[(cb + 0) * HW + p];
            const float v1 = tb[(cb + 1) * HW + p];
            const float v2 = tb[(cb + 2) * HW + p];
            const float v3 = tb[(cb + 3) * HW + p];
            a0 += v0 + v1 + v2 + v3;                // channel sum (argmax)
            a1 = fmaf(v3, w20[3], fmaf(v2, w20[2], fmaf(v1, w20[1], fmaf(v0, w20[0], a1))));
            a2 = fmaf(v3, w21[3], fmaf(v2, w21[2], fmaf(v1, w21[1], fmaf(v0, w21[0], a2))));
            a3 = fmaf(v3, w10[3], fmaf(v2, w10[2], fmaf(v1, w10[1], fmaf(v0, w10[0], a3))));
            a4 = fmaf(v3, w11[3], fmaf(v2, w11[2], fmaf(v1, w11[1], fmaf(v0, w11[0], a4))));
        }
        __syncthreads();                            // slot safe to overwrite
        cslot = (cslot + 1 == RING) ? 0 : cslot + 1;
    }

    // ---- epilogue scratch aliased into the (now dead) ring ----
    float* part = ring;                        // [8][5][49] = 1960 floats
    float* red  = ring + NSTRIPES * 5 * HW;    // [5][49]    = 245 floats
    float* gapa = red + 5 * HW;                // [49]

    if (p < HW) {
        float* pp = part + (stripe * 5) * HW + p;
        pp[0 * HW] = a0;
        pp[1 * HW] = a1;
        pp[2 * HW] = a2;
        pp[3 * HW] = a3;
        pp[4 * HW] = a4;
    }
    __syncthreads();

    if (tid < HW) {
#pragma unroll
        for (int a = 0; a < 5; ++a) {
            float s = 0.f;
#pragma unroll
            for (int k = 0; k < NSTRIPES; ++k) s += part[(k * 5 + a) * HW + tid];
            red[a * HW + tid] = s;
        }
    }
    __syncthreads();

    // ---- argmax over positions (first-max like jnp.argmax) ----
    if (tid == 0) {
        float best = red[0];
        int   bi   = 0;
        for (int q = 1; q < HW; ++q) {
            const float v = red[q];
            if (v > best) { best = v; bi = q; }
        }
        s_idx = bi;
    }
    __syncthreads();
    const int idx = s_idx;

    // ---- polar epilogue ----
    float dist_loss = 0.f, gap = 0.f;
    if (tid < HW) {
        const int q = tid;
        float pred0, pred1;
        if (q == idx) {
            pred0 = 0.f; pred1 = 0.f;          // pred.at[bidx, idx].set(0)
        } else {
            pred0 = fmaxf(0.f, red[1 * HW + q] + red[3 * HW + idx] + bias[0]);
            pred1 = fmaxf(0.f, red[2 * HW + q] + red[4 * HW + idx] + bias[1]);
        }
        const float di = (float)(q / SZ - idx / SZ) / (float)SZ;
        const float dj = (float)(q % SZ - idx % SZ) / (float)SZ;
        const float rel_dist  = sqrtf(di * di + dj * dj);
        const float rel_angle = (atan2f(dj, di) / (float)M_PI + 1.f) * 0.5f;
        const float d0 = pred0 - rel_dist;
        dist_loss = d0 * d0;
        gap = pred1 - rel_angle;
        if (gap < 0.f) gap += 1.f;
        gapa[q] = gap;
    }
    __syncthreads();

    if (tid < HW) {
        float gm = 0.f;
        for (int q = 0; q < HW; ++q) gm += gapa[q];   // fixed order, deterministic
        gm *= (1.f / (float)HW);
        const float g = gap - gm;
        out[(size_t)n * HW + tid] = dist_loss + g * g;
    }
}

extern "C" void kernel_launch(void* const* d_in, const int* in_sizes, int n_in,
                              void* d_out, int out_size, void* d_ws, size_t ws_size,
                              hipStream_t stream) {
    const float* x    = (const float*)d_in[0];
    const float* Wlin = (const float*)d_in[1];
    const float* b    = (const float*)d_in[2];
    float* out        = (float*)d_out;
    (void)in_sizes; (void)n_in; (void)out_size; (void)d_ws; (void)ws_size;

    hipLaunchKernelGGL(rpcp_kernel, dim3(N_BATCH), dim3(NTHREADS), 0, stream,
                       x, Wlin, b, out);
}